// ToM2C_single_25632364822770
// MI455X (gfx1250) — compile-verified
//
#include <hip/hip_runtime.h>

// PropNet edge forward, fused for MI455X (gfx1250, wave32).
// All matmuls via V_WMMA_F32_16X16X4_F32 (exact f32, matches reference).
// Round 2: B-fragments batched into register arrays (no per-WMMA load stalls),
// dual accumulator chains per pass, hot weights staged transposed in LDS so
// fragments are single ds_load_b64.
//
// Shapes: B=4 N=64 R=8 D=32 H=96 E=2.
// Edge rows: B*N*N*R = 131072 of H=96.  Node rows: B*N*R = 2048 of H=96.

#define B_ 4
#define N_ 64
#define R_ 8
#define D_ 32
#define H_ 96
#define E_ 2
#define MN_ (B_*N_*R_)        // 2048 node rows
#define ME_ (B_*N_*N_*R_)     // 131072 edge rows

typedef __attribute__((ext_vector_type(2))) float v2f;
typedef __attribute__((ext_vector_type(8))) float v8f;

__device__ __forceinline__ v8f wmma4(v2f a, v2f b, v8f c) {
  // D = A(16x4) x B(4x16) + C(16x16), all f32, wave32.
  return __builtin_amdgcn_wmma_f32_16x16x4_f32(
      /*neg_a=*/false, a, /*neg_b=*/false, b,
      /*c_mod=*/(short)0, c, /*reuse_a=*/false, /*reuse_b=*/false);
}

// Same-wave LDS RAW/WAR fence (tiles are wave-private; DS per-wave in order).
__device__ __forceinline__ void lds_fence() {
  asm volatile("s_wait_dscnt 0" ::: "memory");
}

// ds_swizzle SWAPX16 (group-of-32: xor=0x10, and=0x1f)
__device__ __forceinline__ float swz_xor16(float x) {
  return __int_as_float(__builtin_amdgcn_ds_swizzle(__float_as_int(x), 0x401f));
}

// A-fragment (f32 16x4): lane(m,half) holds A[m][k0+2*half], A[m][k0+2*half+1]
// -> contiguous pair, loaded as one b64.
template<int NK>
__device__ __forceinline__ void load_afrags(v2f a[NK], const float* row, int half) {
#pragma unroll
  for (int ks = 0; ks < NK; ++ks)
    a[ks] = *(const v2f*)(row + ks*4 + 2*half);
}

__device__ __forceinline__ void zero6(v8f c[6]) {
#pragma unroll
  for (int i = 0; i < 6; ++i) c[i] = (v8f){0,0,0,0,0,0,0,0};
}

// C += A(16 x 4*NK) @ W(4*NK x 96), W in LDS TRANSPOSED: Wt[n*H_ + k] = W[k][n].
// B-fragment pair (W[k0][n], W[k0+1][n]) is contiguous -> single ds_load_b64.
// Two N-tile chains interleaved; fragments batched in chunks of KC (one wait
// per chunk, then a clean run of WMMAs).
template<int NK>
__device__ __forceinline__ void mm2_t(const v2f a[NK], const float* Wt,
                                      int m, int half, v8f c[6]) {
  constexpr int KC = (NK > 12) ? 12 : NK;
#pragma unroll
  for (int ntp = 0; ntp < 3; ++ntp) {
    const float* wp0 = Wt + ((2*ntp + 0)*16 + m)*H_ + 2*half;
    const float* wp1 = wp0 + 16*H_;
    v8f acc0 = c[2*ntp + 0];
    v8f acc1 = c[2*ntp + 1];
#pragma unroll
    for (int kc = 0; kc < NK; kc += KC) {
      v2f b0[KC], b1[KC];
#pragma unroll
      for (int ks = 0; ks < KC; ++ks) {
        b0[ks] = *(const v2f*)(wp0 + (kc + ks)*4);
        b1[ks] = *(const v2f*)(wp1 + (kc + ks)*4);
      }
#pragma unroll
      for (int ks = 0; ks < KC; ++ks) {
        acc0 = wmma4(a[kc + ks], b0[ks], acc0);
        acc1 = wmma4(a[kc + ks], b1[ks], acc1);
      }
    }
    c[2*ntp + 0] = acc0;
    c[2*ntp + 1] = acc1;
  }
}

// Same but W read directly from global, row-major [k][n] (cold/small matmuls).
template<int NK>
__device__ __forceinline__ void mm2_g(const v2f a[NK], const float* __restrict__ W,
                                      int m, int half, v8f c[6]) {
  constexpr int KC = (NK > 12) ? 12 : NK;
#pragma unroll
  for (int ntp = 0; ntp < 3; ++ntp) {
    const float* wp0 = W + (2*half)*H_ + (2*ntp + 0)*16 + m;
    const float* wp1 = wp0 + 16;
    v8f acc0 = c[2*ntp + 0];
    v8f acc1 = c[2*ntp + 1];
#pragma unroll
    for (int kc = 0; kc < NK; kc += KC) {
      v2f b0[KC], b1[KC];
#pragma unroll
      for (int ks = 0; ks < KC; ++ks) {
        b0[ks].x = wp0[(kc + ks)*4*H_];
        b0[ks].y = wp0[(kc + ks)*4*H_ + H_];
        b1[ks].x = wp1[(kc + ks)*4*H_];
        b1[ks].y = wp1[(kc + ks)*4*H_ + H_];
      }
#pragma unroll
      for (int ks = 0; ks < KC; ++ks) {
        acc0 = wmma4(a[kc + ks], b0[ks], acc0);
        acc1 = wmma4(a[kc + ks], b1[ks], acc1);
      }
    }
    c[2*ntp + 0] = acc0;
    c[2*ntp + 1] = acc1;
  }
}

// Stage a 96x96 row-major weight into LDS transposed (Wt[n][k]); 128 threads.
__device__ __forceinline__ void stage_wt(float* dst, const float* __restrict__ W,
                                         int tid) {
  for (int idx = tid*4; idx < H_*H_; idx += 128*4) {
    const float4 w = *(const float4*)(W + idx);   // coalesced b128 read
    const int k = idx / H_;
    const int n = idx - k*H_;                     // H_%4==0 -> no row wrap
    dst[(n + 0)*H_ + k] = w.x;
    dst[(n + 1)*H_ + k] = w.y;
    dst[(n + 2)*H_ + k] = w.z;
    dst[(n + 3)*H_ + k] = w.w;
  }
}

// C/D layout: vgpr v, lane(m,half) -> element (row = v + 8*half, col = nt*16+m).
#define CTILE_FOREACH(...)                              \
  _Pragma("unroll")                                     \
  for (int nt = 0; nt < 6; ++nt) {                      \
    _Pragma("unroll")                                   \
    for (int v = 0; v < 8; ++v) {                       \
      const int col = nt*16 + m;                        \
      const int row = v + 8*half;                       \
      (void)row; (void)col;                             \
      __VA_ARGS__                                       \
    }                                                   \
  }

// ---------------------------------------------------------------------------
// Kernel 1: node_enc = relu(node_rep@ne_W + ne_b); nodeWr/nodeWs = node_rep@Wr/Ws;
//           nEr/nEs = node_enc @ Er0/Es0   (node_effect == node_enc at iter 0)
__global__ void __launch_bounds__(32)
node_prep_kernel(const float* __restrict__ node_rep,
                 const float* __restrict__ ne_W, const float* __restrict__ ne_b,
                 const float* __restrict__ ee_W,
                 const float* __restrict__ ep0_W1,
                 float* __restrict__ node_enc, float* __restrict__ nodeWr,
                 float* __restrict__ nodeWs, float* __restrict__ nEr,
                 float* __restrict__ nEs) {
  __shared__ float tl[16*H_];
  const int lane = threadIdx.x & 31, m = lane & 15, half = lane >> 4;
  const int row0 = blockIdx.x * 16;

  v2f a[8];
  load_afrags<8>(a, node_rep + (row0 + m)*D_, half);
  v8f c[6];

  zero6(c); mm2_g<8>(a, ne_W, m, half, c);
  CTILE_FOREACH(
    float val = fmaxf(c[nt][v] + ne_b[col], 0.0f);
    node_enc[(row0+row)*H_ + col] = val;
    tl[row*H_ + col] = val;
  )

  zero6(c); mm2_g<8>(a, ee_W, m, half, c);             // Wr = ee_W[0:D]
  CTILE_FOREACH( nodeWr[(row0+row)*H_ + col] = c[nt][v]; )

  zero6(c); mm2_g<8>(a, ee_W + D_*H_, m, half, c);     // Ws = ee_W[D:2D]
  CTILE_FOREACH( nodeWs[(row0+row)*H_ + col] = c[nt][v]; )

  lds_fence();
  v2f a2[24];
  load_afrags<24>(a2, tl + m*H_, half);                // node_enc, A-layout

  zero6(c); mm2_g<24>(a2, ep0_W1, m, half, c);         // Er0
  CTILE_FOREACH( nEr[(row0+row)*H_ + col] = c[nt][v]; )

  zero6(c); mm2_g<24>(a2, ep0_W1 + H_*H_, m, half, c); // Es0
  CTILE_FOREACH( nEs[(row0+row)*H_ + col] = c[nt][v]; )
}

// ---------------------------------------------------------------------------
// Kernel 2: edge pass, iteration 0. grid = B*N, block = 128 (4 waves).
// Ee/W2 staged transposed in LDS. agg = sum_j edge_effect, deterministic.
__global__ void __launch_bounds__(128)
edge0_kernel(const float* __restrict__ nodeWr, const float* __restrict__ nodeWs,
             const float* __restrict__ ee_b,
             const float* __restrict__ nEr, const float* __restrict__ nEs,
             const float* __restrict__ ep_W1, const float* __restrict__ ep_b1,
             const float* __restrict__ ep_W2, const float* __restrict__ ep_b2,
             float* __restrict__ EE, float* __restrict__ agg) {
  __shared__ float sEe[H_*H_];         // 36 KB, transposed [n][k]
  __shared__ float sW2[H_*H_];         // 36 KB, transposed [n][k]
  __shared__ float tl[4][16*H_];       // 24 KB, per-wave transpose tiles
  __shared__ float aggl[4][R_*H_];     // 12 KB

  const int wave = threadIdx.x >> 5, lane = threadIdx.x & 31;
  const int m = lane & 15, half = lane >> 4;
  const int bi = blockIdx.x, b = bi / N_;

  stage_wt(sEe, ep_W1 + 2*H_*H_, threadIdx.x);         // Ee = W1[2H:3H]
  stage_wt(sW2, ep_W2, threadIdx.x);
  __syncthreads();

  const float* wr_row = nodeWr + (bi*R_ + (m & 7))*H_; // depends on r(m) only
  const float* er_base = nEr + bi*R_*H_;
  float* tw = tl[wave];
  v8f aggc[6]; zero6(aggc);

  for (int tile = wave; tile < 32; tile += 4) {
    const int j0 = 2*tile;
    const float* ws_row = nodeWs + ((b*N_ + j0 + (m >> 3))*R_ + (m & 7))*H_;

    // A = edge_enc rows (recompute: relu(nodeWr_i + nodeWs_j + ee_b))
    v2f a[24];
#pragma unroll
    for (int ks = 0; ks < 24; ++ks) {
      const int k = ks*4 + 2*half;
      a[ks].x = fmaxf(wr_row[k]   + ws_row[k]   + ee_b[k],   0.0f);
      a[ks].y = fmaxf(wr_row[k+1] + ws_row[k+1] + ee_b[k+1], 0.0f);
    }
    v8f c[6]; zero6(c);
    mm2_t<24>(a, sEe, m, half, c);

    // t = relu(c + nEr[i,r] + nEs[j,r] + b1): row=v+8*half => r=v, j=j0+half
    const float* es_base = nEs + (size_t)(b*N_ + j0 + half)*R_*H_;
    CTILE_FOREACH(
      float val = c[nt][v] + er_base[v*H_ + col] + es_base[v*H_ + col] + ep_b1[col];
      tw[row*H_ + col] = fmaxf(val, 0.0f);
    )
    lds_fence();
    load_afrags<24>(a, tw + m*H_, half);               // transpose C->A via LDS
    lds_fence();

    zero6(c);
    mm2_t<24>(a, sW2, m, half, c);
    const size_t rowbase = (size_t)bi*512 + (size_t)tile*16;
    CTILE_FOREACH(
      float val = fmaxf(c[nt][v] + ep_b2[col], 0.0f);  // edge_effect iter0
      EE[(rowbase + row)*H_ + col] = val;
      aggc[nt][v] += val;                              // r == v
    )
  }

  // Deterministic sum over j: combine lane halves, then 4 waves via LDS.
  float* aw = aggl[wave];
#pragma unroll
  for (int nt = 0; nt < 6; ++nt)
#pragma unroll
    for (int v = 0; v < 8; ++v) {
      float s = aggc[nt][v];
      s += swz_xor16(s);
      if (half == 0) aw[v*H_ + nt*16 + m] = s;
    }
  __syncthreads();
  for (int idx = threadIdx.x; idx < R_*H_; idx += 128)
    agg[bi*R_*H_ + idx] = aggl[0][idx] + aggl[1][idx] + aggl[2][idx] + aggl[3][idx];
}

// ---------------------------------------------------------------------------
// Kernel 3: node update (iter 0) fused with iter-1 projections.
__global__ void __launch_bounds__(32)
node_pass_kernel(const float* __restrict__ node_enc, const float* __restrict__ agg,
                 const float* __restrict__ np_W1, const float* __restrict__ np_b1,
                 const float* __restrict__ np_W2, const float* __restrict__ np_b2,
                 const float* __restrict__ ep1_W1,
                 float* __restrict__ nEr, float* __restrict__ nEs) {
  __shared__ float tl[16*H_];
  const int lane = threadIdx.x & 31, m = lane & 15, half = lane >> 4;
  const int row0 = blockIdx.x * 16;

  v2f a[24];
  v8f c[6]; zero6(c);
  load_afrags<24>(a, node_enc + (row0 + m)*H_, half);
  mm2_g<24>(a, np_W1, m, half, c);                     // Wn1 @ node_enc
  mm2_g<24>(a, np_W1 + H_*H_, m, half, c);             // Wn2 @ node_effect(=enc)
  load_afrags<24>(a, agg + (row0 + m)*H_, half);
  mm2_g<24>(a, np_W1 + 2*H_*H_, m, half, c);           // Wn3 @ agg

  CTILE_FOREACH( tl[row*H_ + col] = fmaxf(c[nt][v] + np_b1[col], 0.0f); )
  lds_fence();
  load_afrags<24>(a, tl + m*H_, half);
  lds_fence();

  zero6(c); mm2_g<24>(a, np_W2, m, half, c);
  CTILE_FOREACH( tl[row*H_ + col] = fmaxf(c[nt][v] + np_b2[col], 0.0f); )
  lds_fence();
  load_afrags<24>(a, tl + m*H_, half);                 // node_effect, A-layout

  zero6(c); mm2_g<24>(a, ep1_W1, m, half, c);          // Er1
  CTILE_FOREACH( nEr[(row0+row)*H_ + col] = c[nt][v]; )
  zero6(c); mm2_g<24>(a, ep1_W1 + H_*H_, m, half, c);  // Es1
  CTILE_FOREACH( nEs[(row0+row)*H_ + col] = c[nt][v]; )
}

// ---------------------------------------------------------------------------
// Kernel 4: edge pass iteration 1 fused with prediction head.
__global__ void __launch_bounds__(128)
edge1_kernel(const float* __restrict__ EE,
             const float* __restrict__ nodeWr, const float* __restrict__ nodeWs,
             const float* __restrict__ ee_b,
             const float* __restrict__ nEr, const float* __restrict__ nEs,
             const float* __restrict__ ep_W1, const float* __restrict__ ep_b1,
             const float* __restrict__ ep_W2, const float* __restrict__ ep_b2,
             const float* __restrict__ pred_W1, const float* __restrict__ pred_b1,
             const float* __restrict__ pred_W2, const float* __restrict__ pred_b2,
             float* __restrict__ out) {
  __shared__ float sEe[H_*H_];         // 36 KB transposed
  __shared__ float sW2[H_*H_];         // 36 KB transposed
  __shared__ float tl[4][16*H_];       // 24 KB
  __shared__ float pwl[H_*E_];         // 768 B

  const int wave = threadIdx.x >> 5, lane = threadIdx.x & 31;
  const int m = lane & 15, half = lane >> 4;
  const int bi = blockIdx.x, b = bi / N_;
  const float* Pe = pred_W1;                           // pred_W1[0:H]
  const float* Pc = pred_W1 + H_*H_;                   // pred_W1[H:2H]

  // warm L2/L0 for the globally-read pred weights (global_prefetch_b8)
  for (int idx = threadIdx.x*32; idx < 2*H_*H_; idx += 128*32)
    __builtin_prefetch(pred_W1 + idx, 0, 3);

  stage_wt(sEe, ep_W1 + 2*H_*H_, threadIdx.x);
  stage_wt(sW2, ep_W2, threadIdx.x);
  for (int idx = threadIdx.x; idx < H_*E_; idx += 128) pwl[idx] = pred_W2[idx];
  __syncthreads();

  const float* wr_row = nodeWr + (bi*R_ + (m & 7))*H_;
  const float* er_base = nEr + bi*R_*H_;
  const float pb2 = pred_b2[half];
  float* tw = tl[wave];

  for (int tile = wave; tile < 32; tile += 4) {
    const int j0 = 2*tile;
    const size_t rowbase = (size_t)bi*512 + (size_t)tile*16;

    v2f a[24];
    load_afrags<24>(a, EE + (rowbase + m)*H_, half);   // edge_effect (iter0)
    v8f c[6]; zero6(c);
    mm2_t<24>(a, sEe, m, half, c);

    const float* es_base = nEs + (size_t)(b*N_ + j0 + half)*R_*H_;
    CTILE_FOREACH(
      float val = c[nt][v] + er_base[v*H_ + col] + es_base[v*H_ + col] + ep_b1[col];
      tw[row*H_ + col] = fmaxf(val, 0.0f);
    )
    lds_fence();
    load_afrags<24>(a, tw + m*H_, half);
    lds_fence();

    zero6(c); mm2_t<24>(a, sW2, m, half, c);
    CTILE_FOREACH( tw[row*H_ + col] = fmaxf(c[nt][v] + ep_b2[col], 0.0f); ) // final EE
    lds_fence();
    load_afrags<24>(a, tw + m*H_, half);
    lds_fence();

    zero6(c);
    mm2_g<24>(a, Pe, m, half, c);                      // edge_effect @ Pe
    {
      // recompute edge_enc in registers, accumulate @ Pc into same C
      const float* ws_row = nodeWs + ((b*N_ + j0 + (m >> 3))*R_ + (m & 7))*H_;
      v2f ae[24];
#pragma unroll
      for (int ks = 0; ks < 24; ++ks) {
        const int k = ks*4 + 2*half;
        ae[ks].x = fmaxf(wr_row[k]   + ws_row[k]   + ee_b[k],   0.0f);
        ae[ks].y = fmaxf(wr_row[k+1] + ws_row[k+1] + ee_b[k+1], 0.0f);
      }
      mm2_g<24>(ae, Pc, m, half, c);
    }
    CTILE_FOREACH( tw[row*H_ + col] = fmaxf(c[nt][v] + pred_b1[col], 0.0f); ) // t2
    lds_fence();

    // 96 -> E=2 head: lane(m,half) computes out[row=m][e=half]
    {
      float s = pb2;
#pragma unroll 8
      for (int colx = 0; colx < H_; ++colx)
        s += tw[m*H_ + colx] * pwl[colx*E_ + half];
      out[(rowbase + m)*E_ + half] = s;
    }
    lds_fence();  // protect tw reads before next tile overwrites
  }
}

// ---------------------------------------------------------------------------
extern "C" void kernel_launch(void* const* d_in, const int* in_sizes, int n_in,
                              void* d_out, int out_size, void* d_ws, size_t ws_size,
                              hipStream_t stream) {
  const float* node_rep = (const float*)d_in[0];
  const float* ne_W     = (const float*)d_in[1];
  const float* ne_b     = (const float*)d_in[2];
  const float* ee_W     = (const float*)d_in[3];
  const float* ee_b     = (const float*)d_in[4];
  const float* np_W1    = (const float*)d_in[5];
  const float* np_b1    = (const float*)d_in[6];
  const float* np_W2    = (const float*)d_in[7];
  const float* np_b2    = (const float*)d_in[8];
  const float* ep0_W1   = (const float*)d_in[9];
  const float* ep0_b1   = (const float*)d_in[10];
  const float* ep0_W2   = (const float*)d_in[11];
  const float* ep0_b2   = (const float*)d_in[12];
  const float* ep1_W1   = (const float*)d_in[13];
  const float* ep1_b1   = (const float*)d_in[14];
  const float* ep1_W2   = (const float*)d_in[15];
  const float* ep1_b2   = (const float*)d_in[16];
  const float* pred_W1  = (const float*)d_in[17];
  const float* pred_b1  = (const float*)d_in[18];
  const float* pred_W2  = (const float*)d_in[19];
  const float* pred_b2  = (const float*)d_in[20];
  (void)in_sizes; (void)n_in; (void)out_size; (void)ws_size;

  // Workspace: 6 node-grid buffers (2048x96 each) + iter-0 edge_effect (131072x96)
  float* ws       = (float*)d_ws;
  float* node_enc = ws; ws += MN_*H_;
  float* nodeWr   = ws; ws += MN_*H_;
  float* nodeWs   = ws; ws += MN_*H_;
  float* nEr      = ws; ws += MN_*H_;
  float* nEs      = ws; ws += MN_*H_;
  float* agg      = ws; ws += MN_*H_;
  float* EE       = ws;                // 12582912 floats (~50 MB)

  node_prep_kernel<<<MN_/16, 32, 0, stream>>>(node_rep, ne_W, ne_b, ee_W, ep0_W1,
                                              node_enc, nodeWr, nodeWs, nEr, nEs);
  edge0_kernel<<<B_*N_, 128, 0, stream>>>(nodeWr, nodeWs, ee_b, nEr, nEs,
                                          ep0_W1, ep0_b1, ep0_W2, ep0_b2, EE, agg);
  node_pass_kernel<<<MN_/16, 32, 0, stream>>>(node_enc, agg, np_W1, np_b1,
                                              np_W2, np_b2, ep1_W1, nEr, nEs);
  edge1_kernel<<<B_*N_, 128, 0, stream>>>(EE, nodeWr, nodeWs, ee_b, nEr, nEs,
                                          ep1_W1, ep1_b1, ep1_W2, ep1_b2,
                                          pred_W1, pred_b1, pred_W2, pred_b2,
                                          (float*)d_out);
}